// ConventionalGNN_12068858102068
// MI455X (gfx1250) — compile-verified
//
#include <hip/hip_runtime.h>

typedef __attribute__((ext_vector_type(2))) float v2f;
typedef __attribute__((ext_vector_type(8))) float v8f;

#define GN_N 100000
#define GN_E 1600000
#define GN_DH 128
#define GN_EPS 1e-5f

__device__ __forceinline__ void atomicAddF(float* p, float v) {
    __hip_atomic_fetch_add(p, v, __ATOMIC_RELAXED, __HIP_MEMORY_SCOPE_AGENT);
}

// ---------------- degree / dinv ----------------
__global__ void degree_kernel(const int* __restrict__ dst, float* __restrict__ deg, int e_cnt) {
    int e = blockIdx.x * blockDim.x + threadIdx.x;
    if (e < e_cnt) atomicAddF(&deg[dst[e]], 1.0f);
}

__global__ void dinv_kernel(float* __restrict__ deg, int n) {
    int i = blockIdx.x * blockDim.x + threadIdx.x;
    if (i < n) deg[i] = rsqrtf(deg[i] + 1.0f);
}

// ---------------- layer-0 GEMM: X[N,4] @ W0[4,128], single WMMA per tile ----------------
__global__ __launch_bounds__(256) void gemm_k4_wmma(const float* __restrict__ X,
                                                    const float* __restrict__ W,
                                                    float* __restrict__ H) {
    int wave = threadIdx.x >> 5;
    int lane = threadIdx.x & 31;
    int tile = blockIdx.x * 8 + wave;                   // 50000 tiles total
    int mt = tile >> 3, nt = tile & 7;
    int m0 = mt * 16, n0 = nt * 16;
    int ln = lane & 15;
    int kh = (lane >> 4) * 2;                           // 0 (lanes 0-15) / 2 (lanes 16-31)

    v2f a, b;
    const float* xr = X + (size_t)(m0 + ln) * 4 + kh;   // row m, cols kh,kh+1 (contiguous)
    a.x = xr[0];
    a.y = xr[1];
    b.x = W[(size_t)(kh + 0) * GN_DH + n0 + ln];        // B: K across VGPRs, N across lanes
    b.y = W[(size_t)(kh + 1) * GN_DH + n0 + ln];

    v8f c = {};
    c = __builtin_amdgcn_wmma_f32_16x16x4_f32(false, a, false, b, (short)0, c, false, false);

    float* out = H + (size_t)(m0 + (lane >> 4) * 8) * GN_DH + n0 + ln;
#pragma unroll
    for (int v = 0; v < 8; ++v) out[(size_t)v * GN_DH] = c[v];
}

// ---------------- hidden GEMM: A[N,128] @ W[128,128], fp32 WMMA, fully unrolled K ----------------
__global__ __launch_bounds__(256) void gemm_k128_wmma(const float* __restrict__ A,
                                                      const float* __restrict__ W,
                                                      float* __restrict__ H) {
    int wave = threadIdx.x >> 5;
    int lane = threadIdx.x & 31;
    int tile = blockIdx.x * 8 + wave;    // 8 waves/block share one M tile, cover all 8 N tiles
    int mt = tile >> 3, nt = tile & 7;
    int m0 = mt * 16, n0 = nt * 16;
    int ln = lane & 15;
    int kh = (lane >> 4) * 2;

    const float2* arow = reinterpret_cast<const float2*>(A + (size_t)(m0 + ln) * GN_DH + kh);
    const float* wcol = W + n0 + ln;

    v8f c = {};
#pragma unroll
    for (int kb = 0; kb < 32; ++kb) {
        int k0 = kb * 4 + kh;
        float2 av = arow[kb * 2];                 // cols k0, k0+1 of row m (8B load)
        v2f a, b;
        a.x = av.x;
        a.y = av.y;
        b.x = wcol[(size_t)(k0 + 0) * GN_DH];
        b.y = wcol[(size_t)(k0 + 1) * GN_DH];
        c = __builtin_amdgcn_wmma_f32_16x16x4_f32(false, a, false, b, (short)0, c, false, false);
    }

    float* out = H + (size_t)(m0 + (lane >> 4) * 8) * GN_DH + n0 + ln;
#pragma unroll
    for (int v = 0; v < 8; ++v) out[(size_t)v * GN_DH] = c[v];
}

// ---------------- edge scatter-add: one wave per edge, float4 per lane ----------------
__global__ void edge_aggregate(const int* __restrict__ src, const int* __restrict__ dst,
                               const float* __restrict__ dinv, const float* __restrict__ H,
                               float* __restrict__ AGG, int e_cnt) {
    int wave = threadIdx.x >> 5;
    int lane = threadIdx.x & 31;
    int e = blockIdx.x * 8 + wave;
    if (e >= e_cnt) return;
    int s = src[e], d = dst[e];
    float coef = dinv[s] * dinv[d];
    float4 v = reinterpret_cast<const float4*>(H + (size_t)s * GN_DH)[lane];
    float* out = AGG + (size_t)d * GN_DH + lane * 4;
    atomicAddF(out + 0, v.x * coef);
    atomicAddF(out + 1, v.y * coef);
    atomicAddF(out + 2, v.z * coef);
    atomicAddF(out + 3, v.w * coef);
}

// ---------------- fused self-loop + bias + per-feature sum / sumsq reduction ----------------
__global__ void selfloop_bias_stats(float* __restrict__ Y,        // agg in, y out (in place)
                                    const float* __restrict__ H,  // pre-aggregation features
                                    const float* __restrict__ dinv,
                                    const float* __restrict__ bias,
                                    float* __restrict__ ssum, float* __restrict__ ssq,
                                    int rows) {
    int f = threadIdx.x;            // 128 threads, one per feature
    int r0 = blockIdx.x * 64;
    int r1 = r0 + 64;
    if (r1 > rows) r1 = rows;
    float bf = bias[f];
    float acc = 0.f, acc2 = 0.f;
    for (int r = r0; r < r1; ++r) {
        float di = dinv[r];
        size_t idx = (size_t)r * GN_DH + f;
        float y = Y[idx] + H[idx] * (di * di) + bf;
        Y[idx] = y;
        acc += y;
        acc2 += y * y;
    }
    atomicAddF(ssum + f, acc);
    atomicAddF(ssq + f, acc2);
}

// ---------------- GraphNorm scale/shift from moments ----------------
__global__ void finalize_stats(const float* __restrict__ ssum, const float* __restrict__ ssq,
                               const float* __restrict__ gamma, const float* __restrict__ beta,
                               const float* __restrict__ alpha,
                               float* __restrict__ scale, float* __restrict__ shift, float invN) {
    int f = threadIdx.x;
    float mu = ssum[f] * invN;
    float ex2 = ssq[f] * invN;
    float a = alpha[f];
    // var = E[(x - a*mu)^2] = E[x^2] - 2*a*mu*E[x] + a^2*mu^2
    float var = ex2 - 2.f * a * mu * mu + a * a * mu * mu;
    float sc = gamma[f] * rsqrtf(var + GN_EPS);
    scale[f] = sc;
    shift[f] = beta[f] - sc * a * mu;
}

// ---------------- normalize + ReLU (vectorized float4) ----------------
__global__ void norm_relu(const float* __restrict__ Y, const float* __restrict__ scale,
                          const float* __restrict__ shift, float* __restrict__ O) {
    size_t i = (size_t)blockIdx.x * blockDim.x + threadIdx.x;  // over N*128/4 float4s
    int f = (int)(i & 31) * 4;                                  // 32 float4 per row
    float4 y = reinterpret_cast<const float4*>(Y)[i];
    float4 o;
    o.x = fmaxf(scale[f + 0] * y.x + shift[f + 0], 0.f);
    o.y = fmaxf(scale[f + 1] * y.y + shift[f + 1], 0.f);
    o.z = fmaxf(scale[f + 2] * y.z + shift[f + 2], 0.f);
    o.w = fmaxf(scale[f + 3] * y.w + shift[f + 3], 0.f);
    reinterpret_cast<float4*>(O)[i] = o;
}

extern "C" void kernel_launch(void* const* d_in, const int* in_sizes, int n_in,
                              void* d_out, int out_size, void* d_ws, size_t ws_size,
                              hipStream_t stream) {
    const float* x     = (const float*)d_in[0];
    const int*   ei    = (const int*)d_in[1];   // [2,E]
    const float* W0    = (const float*)d_in[2];
    const float* b0    = (const float*)d_in[3];
    const float* W12   = (const float*)d_in[4]; // [2,128,128]
    const float* b12   = (const float*)d_in[5]; // [2,128]
    const float* gamma = (const float*)d_in[6]; // [3,128]
    const float* beta  = (const float*)d_in[7];
    const float* alpha = (const float*)d_in[8];
    float* out = (float*)d_out;
    (void)in_sizes; (void)n_in; (void)out_size; (void)ws_size;

    const int N = GN_N, E = GN_E, DH = GN_DH;
    const int* src = ei;
    const int* dst = ei + E;

    // workspace layout (floats)
    float* dinv  = (float*)d_ws;          // N (padded to 100096)
    float* ssum  = dinv + 100096;         // 128
    float* ssq   = ssum + 128;            // 128
    float* scale = ssq + 128;             // 128
    float* shift = scale + 128;           // 128
    float* bufA  = shift + 128;           // N*128
    float* bufB  = bufA + (size_t)N * DH; // N*128

    // degrees -> dinv
    hipMemsetAsync(dinv, 0, (size_t)N * sizeof(float), stream);
    degree_kernel<<<(E + 255) / 256, 256, 0, stream>>>(dst, dinv, E);
    dinv_kernel<<<(N + 255) / 256, 256, 0, stream>>>(dinv, N);

    const int tiles = (N / 16) * (DH / 16);   // 50000
    const int gemmBlocks = tiles / 8;         // 6250, 8 waves/block, exact
    const int edgeBlocks = E / 8;             // 200000, 8 edges/block, exact
    const int statBlocks = (N + 63) / 64;     // 1563
    const int nrBlocks = (N * DH / 4) / 256;  // 12500, exact
    const float invN = 1.0f / (float)N;

    float* h  = bufA;   // GEMM output / gather source
    float* ag = bufB;   // aggregation target / normalized output
    for (int layer = 0; layer < 3; ++layer) {
        const float* in = (layer == 0) ? x : ag;  // after swap, `ag` holds the previous
                                                  // layer's GEMM output
        if (layer == 0) {
            gemm_k4_wmma<<<gemmBlocks, 256, 0, stream>>>(x, W0, h);
        } else {
            gemm_k128_wmma<<<gemmBlocks, 256, 0, stream>>>(in, W12 + (size_t)(layer - 1) * DH * DH, h);
        }

        hipMemsetAsync(ag, 0, (size_t)N * DH * sizeof(float), stream);
        hipMemsetAsync(ssum, 0, 2 * DH * sizeof(float), stream);  // ssum + ssq contiguous

        edge_aggregate<<<edgeBlocks, 256, 0, stream>>>(src, dst, dinv, h, ag, E);

        const float* bias = (layer == 0) ? b0 : (b12 + (size_t)(layer - 1) * DH);
        selfloop_bias_stats<<<statBlocks, 128, 0, stream>>>(ag, h, dinv, bias, ssum, ssq, N);
        finalize_stats<<<1, 128, 0, stream>>>(ssum, ssq,
                                              gamma + (size_t)layer * DH,
                                              beta + (size_t)layer * DH,
                                              alpha + (size_t)layer * DH,
                                              scale, shift, invN);

        float* dest = (layer == 2) ? out : ag;  // normalize in place (ag) except final layer
        norm_relu<<<nrBlocks, 256, 0, stream>>>(ag, scale, shift, dest);

        // swap roles: `ag` (normalized output) becomes next GEMM output; `h` feeds next GEMM, then agg target
        float* t = h; h = ag; ag = t;
    }
}